// Exp_Linear_11476152615033
// MI455X (gfx1250) — compile-verified
//
#include <hip/hip_runtime.h>

typedef __attribute__((ext_vector_type(2))) float v2f;
typedef __attribute__((ext_vector_type(8))) float v8f;

#define N_NODES 100000
#define N_EDGES 800000
#define DIM     64
#define NHEAD   4

__device__ __forceinline__ v2f ld2(const float* p) { return *(const v2f*)p; }

// ---------------- QKV GEMM: Q/K/V = x @ W^T via V_WMMA_F32_16X16X4_F32 ---------------
__global__ __launch_bounds__(256) void qkv_kernel(
    const float* __restrict__ x,
    const float* __restrict__ Wq, const float* __restrict__ Wk, const float* __restrict__ Wv,
    float* __restrict__ Q, float* __restrict__ K, float* __restrict__ V)
{
    const int lane = threadIdx.x & 31;
    const int wave = threadIdx.x >> 5;
    const int tile = blockIdx.x * 8 + wave;          // 16-row tile of x
    const int ntiles = N_NODES / 16;                 // 6250
    if (tile >= ntiles) return;
    const int half = lane >> 4;                      // 0: K even pair, 1: K odd pair
    const int lm   = lane & 15;
    const int r0   = tile * 16;

    // A frags: lane (half,lm) holds x[r0+lm, 4s + 2*half .. +1] for s=0..15
    v2f a[16];
    {
        const float* arow = x + (size_t)(r0 + lm) * DIM + 2 * half;
#pragma unroll
        for (int s = 0; s < 16; ++s) a[s] = ld2(arow + 4 * s);
    }

    const float* Ws[3] = {Wq, Wk, Wv};
    float*       Os[3] = {Q,  K,  V};
#pragma unroll
    for (int m = 0; m < 3; ++m) {
        const float* W = Ws[m];
        float*       O = Os[m];
#pragma unroll
        for (int t = 0; t < 4; ++t) {
            v8f acc = {};
            const int d = 16 * t + lm;               // output column (n index)
#pragma unroll
            for (int s = 0; s < 16; ++s) {
                // B[j, d] = W[d, j]; lane needs j0 = 4s + 2*half, j0+1
                v2f b = ld2(W + d * DIM + 4 * s + 2 * half);
                acc = __builtin_amdgcn_wmma_f32_16x16x4_f32(
                          false, a[s], false, b, (short)0, acc, false, false);
            }
            // C/D layout: VGPR r holds row r (lanes 0-15) / r+8 (lanes 16-31), col = lm+16t
#pragma unroll
            for (int r = 0; r < 8; ++r)
                O[(size_t)(r0 + r + 8 * half) * DIM + d] = acc[r];
        }
    }
}

// -------- Fused edge kernel: E-tile via WMMA (LDS-staged) + attention + scatter --------
__global__ __launch_bounds__(256) void edge_kernel(
    const float* __restrict__ edge_attr, const float* __restrict__ We,
    const long long* __restrict__ edge_index,
    const float* __restrict__ Q, const float* __restrict__ K, const float* __restrict__ V,
    float* __restrict__ wV, float* __restrict__ Z)
{
    __shared__ float etile_lds[8][16 * 66];          // per-wave 16x64 E tile, stride 66
    const int lane = threadIdx.x & 31;
    const int wave = threadIdx.x >> 5;
    const int tile = blockIdx.x * 8 + wave;          // 16-edge tile
    const int ntiles = N_EDGES / 16;                 // 50000
    if (tile >= ntiles) return;
    const int half = lane >> 4;
    const int lm   = lane & 15;
    const int e0   = tile * 16;
    float* Et = etile_lds[wave];

    // ---- E[e0:e0+16, :] = edge_attr @ We^T (WMMA f32 16x16x4, K=64) ----
    v2f a[16];
    {
        const float* arow = edge_attr + (size_t)(e0 + lm) * DIM + 2 * half;
#pragma unroll
        for (int s = 0; s < 16; ++s) a[s] = ld2(arow + 4 * s);
    }
#pragma unroll
    for (int t = 0; t < 4; ++t) {
        v8f acc = {};
        const int d = 16 * t + lm;
#pragma unroll
        for (int s = 0; s < 16; ++s) {
            v2f b = ld2(We + d * DIM + 4 * s + 2 * half);
            acc = __builtin_amdgcn_wmma_f32_16x16x4_f32(
                      false, a[s], false, b, (short)0, acc, false, false);
        }
#pragma unroll
        for (int r = 0; r < 8; ++r)
            Et[(r + 8 * half) * 66 + d] = acc[r];    // stride-66 pad: conflict-free
    }
    // Same-wave LDS RAW: DS ops are in-order per wave; compiler inserts s_wait_dscnt.

    // ---- attention + scatter: one edge per iteration, 2 dims per lane ----
    const int head = lane >> 3;                      // dims 2*lane..2*lane+1 -> head lane/8
    for (int i = 0; i < 16; ++i) {
        const long long src = edge_index[e0 + i];            // row 0 of edge_index
        const long long dst = edge_index[N_EDGES + e0 + i];  // row 1
        v2f kv = ld2(K + (size_t)src * DIM + 2 * lane);
        v2f qv = ld2(Q + (size_t)dst * DIM + 2 * lane);
        v2f vv = ld2(V + (size_t)src * DIM + 2 * lane);
        v2f ev = ld2(&Et[i * 66 + 2 * lane]);
        // score partial: K*Q*E / sqrt(16)
        float part = (kv.x * qv.x * ev.x + kv.y * qv.y * ev.y) * 0.25f;
        // butterfly reduce within 8-lane head group (16 dims)
        part += __shfl_xor(part, 1, 32);
        part += __shfl_xor(part, 2, 32);
        part += __shfl_xor(part, 4, 32);
        const float sc = __expf(fminf(fmaxf(part, -5.0f), 5.0f));
        float* wp = wV + (size_t)dst * DIM + 2 * lane;
        atomicAdd(wp,     vv.x * sc);
        atomicAdd(wp + 1, vv.y * sc);
        if ((lane & 7) == 0)
            atomicAdd(Z + (size_t)dst * NHEAD + head, sc);
    }
}

// ---------------- helpers ----------------
__global__ void zero_kernel(float* __restrict__ p, size_t n) {
    size_t i = (size_t)blockIdx.x * blockDim.x + threadIdx.x;
    if (i < n) p[i] = 0.0f;
}

__global__ void norm_kernel(const float* __restrict__ wV, const float* __restrict__ Z,
                            float* __restrict__ out) {
    size_t i = (size_t)blockIdx.x * blockDim.x + threadIdx.x;
    if (i < (size_t)N_NODES * DIM) {
        size_t n = i >> 6;
        int    d = (int)(i & 63);
        out[i] = wV[i] / (Z[n * NHEAD + (d >> 4)] + 1e-6f);
    }
}

extern "C" void kernel_launch(void* const* d_in, const int* in_sizes, int n_in,
                              void* d_out, int out_size, void* d_ws, size_t ws_size,
                              hipStream_t stream) {
    const float*      x         = (const float*)d_in[0];
    const float*      edge_attr = (const float*)d_in[1];
    const float*      Wq        = (const float*)d_in[2];
    const float*      Wk        = (const float*)d_in[3];
    const float*      We        = (const float*)d_in[4];
    const float*      Wv        = (const float*)d_in[5];
    const long long*  eidx      = (const long long*)d_in[6];
    float*            out       = (float*)d_out;

    float* ws = (float*)d_ws;
    float* Q  = ws;
    float* K  = Q  + (size_t)N_NODES * DIM;
    float* V  = K  + (size_t)N_NODES * DIM;
    float* wV = V  + (size_t)N_NODES * DIM;
    float* Zs = wV + (size_t)N_NODES * DIM;   // [N_NODES * NHEAD], contiguous after wV

    const size_t nz = (size_t)N_NODES * DIM + (size_t)N_NODES * NHEAD;
    zero_kernel<<<(unsigned)((nz + 255) / 256), 256, 0, stream>>>(wV, nz);

    qkv_kernel<<<(N_NODES / 16 + 7) / 8, 256, 0, stream>>>(x, Wq, Wk, Wv, Q, K, V);

    edge_kernel<<<(N_EDGES / 16) / 8, 256, 0, stream>>>(edge_attr, We, eidx, Q, K, V, wV, Zs);

    const size_t no = (size_t)N_NODES * DIM;
    norm_kernel<<<(unsigned)((no + 255) / 256), 256, 0, stream>>>(wV, Zs, out);
}